// SGCMem_62689342652834
// MI455X (gfx1250) — compile-verified
//
#include <hip/hip_runtime.h>

typedef __attribute__((ext_vector_type(2))) float v2f;
typedef __attribute__((ext_vector_type(8))) float v8f;

#define N_NODES 50000
#define N_EDGES 1600000
#define IN_CH   512
#define OUT_CH  128

// ---------------------------------------------------------------------------
// W transpose: Wt[n][k] = W[k][n]  (makes B-fragment loads contiguous float2)
// ---------------------------------------------------------------------------
__global__ void k_transpose_w(const float* __restrict__ W, float* __restrict__ Wt) {
    int idx = blockIdx.x * blockDim.x + threadIdx.x;   // 0 .. 65535
    int k = idx >> 7;     // / OUT_CH
    int n = idx & 127;    // % OUT_CH
    Wt[(size_t)n * IN_CH + k] = W[idx];
}

// ---------------------------------------------------------------------------
// GEMM: h = x @ W + b  via V_WMMA_F32_16X16X4_F32 (fp32, matches reference)
// grid = 3125 blocks (16 rows each), 128 threads = 4 waves,
// wave w computes the 16x32 output slab at columns [w*32, w*32+32).
// A frag (16x4):  lane m=lane&15, VGPR pair holds K = 2*(lane>>4) + {0,1}
// B frag (4x16):  lane n=lane&15, VGPR pair holds K = 2*(lane>>4) + {0,1}
// C/D (16x16):    VGPR r: lanes0-15 -> M=r, lanes16-31 -> M=r+8, N=lane&15
// ---------------------------------------------------------------------------
__global__ void __launch_bounds__(128)
k_gemm_wmma(const float* __restrict__ x, const float* __restrict__ Wt,
            const float* __restrict__ bias, float* __restrict__ h) {
    const int lane  = threadIdx.x & 31;
    const int wv    = threadIdx.x >> 5;          // 0..3
    const int row0  = blockIdx.x * 16;
    const int ncol0 = wv * 32;

    const int mn = lane & 15;          // M for A frag, N for B frag
    const int kg = (lane >> 4) << 1;   // K sub-offset within the x4 step: 0 or 2

    const float* ap  = x  + (size_t)(row0  + mn) * IN_CH + kg;
    const float* bp0 = Wt + (size_t)(ncol0 + mn) * IN_CH + kg;
    const float* bp1 = bp0 + (size_t)16 * IN_CH;

    v8f acc0 = {0.f, 0.f, 0.f, 0.f, 0.f, 0.f, 0.f, 0.f};
    v8f acc1 = acc0;

#pragma unroll 8
    for (int kb = 0; kb < IN_CH; kb += 4) {
        v2f a  = *(const v2f*)(ap  + kb);
        v2f b0 = *(const v2f*)(bp0 + kb);
        v2f b1 = *(const v2f*)(bp1 + kb);
        acc0 = __builtin_amdgcn_wmma_f32_16x16x4_f32(false, a, false, b0,
                                                     (short)0, acc0, false, false);
        acc1 = __builtin_amdgcn_wmma_f32_16x16x4_f32(false, a, false, b1,
                                                     (short)0, acc1, false, false);
    }

    const int   mhalf = lane >> 4;
    const float bia0  = bias[ncol0 + mn];
    const float bia1  = bias[ncol0 + 16 + mn];
#pragma unroll
    for (int r = 0; r < 8; ++r) {
        int mm = r + 8 * mhalf;
        float* o = h + (size_t)(row0 + mm) * OUT_CH + ncol0 + mn;
        o[0]  = acc0[r] + bia0;
        o[16] = acc1[r] + bia1;
    }
}

// ---------------------------------------------------------------------------
// Degree / normalization
// ---------------------------------------------------------------------------
__global__ void k_deg_init(float* __restrict__ deg) {
    int i = blockIdx.x * blockDim.x + threadIdx.x;
    if (i < N_NODES) deg[i] = 1.0f;              // self-loop contribution
}

__global__ void k_deg_scatter(const int* __restrict__ col, float* __restrict__ deg) {
    int e = blockIdx.x * blockDim.x + threadIdx.x;
    if (e < N_EDGES) atomicAdd(&deg[col[e]], 1.0f);
}

__global__ void k_dinv(float* __restrict__ deg) {
    int i = blockIdx.x * blockDim.x + threadIdx.x;
    if (i < N_NODES) {
        float d = deg[i];
        deg[i] = (d > 0.0f) ? rsqrtf(d) : 0.0f;  // in place: deg -> dinv
    }
}

// ---------------------------------------------------------------------------
// Propagation hop.  Self-loop term as init (no atomics, also clears poison):
//   dst[i,:] = dinv[i]^2 * src[i,:]
// then edge scatter:  dst[col_e,:] += dinv[row_e]*dinv[col_e] * src[row_e,:]
// ---------------------------------------------------------------------------
__global__ void k_hop_init(const float* __restrict__ src, const float* __restrict__ dinv,
                           float* __restrict__ dst) {
    int i = blockIdx.x * blockDim.x + threadIdx.x;    // 6,400,000 total
    if (i < N_NODES * OUT_CH) {
        float di = dinv[i >> 7];                      // i / OUT_CH
        dst[i] = di * di * src[i];
    }
}

// one wave per edge: each lane handles 4 channels (32 lanes * 4 = 128)
__global__ void __launch_bounds__(256)
k_hop_scatter(const int* __restrict__ row, const int* __restrict__ col,
              const float* __restrict__ dinv, const float* __restrict__ src,
              float* dst) {
    int e = blockIdx.x * (blockDim.x >> 5) + (threadIdx.x >> 5);
    if (e >= N_EDGES) return;
    int lane = threadIdx.x & 31;

    int r = row[e];
    int c = col[e];
    float w = dinv[r] * dinv[c];

    const float4* s = (const float4*)(src + (size_t)r * OUT_CH);
    float4 v = s[lane];                               // single b128 gather from L2
    float* d = dst + (size_t)c * OUT_CH + lane * 4;
    atomicAdd(d + 0, w * v.x);
    atomicAdd(d + 1, w * v.y);
    atomicAdd(d + 2, w * v.z);
    atomicAdd(d + 3, w * v.w);
}

// ---------------------------------------------------------------------------
// Launch
// ---------------------------------------------------------------------------
extern "C" void kernel_launch(void* const* d_in, const int* in_sizes, int n_in,
                              void* d_out, int out_size, void* d_ws, size_t ws_size,
                              hipStream_t stream) {
    const float* x    = (const float*)d_in[0];
    const float* W    = (const float*)d_in[1];
    const float* bias = (const float*)d_in[2];
    const int*   ei   = (const int*)d_in[3];
    const int*   row  = ei;
    const int*   col  = ei + N_EDGES;

    // workspace layout
    float* bufA = (float*)d_ws;                         // 6,400,000 f (25.6 MB)
    float* dinv = bufA + (size_t)N_NODES * OUT_CH;      //    50,000 f
    float* Wt   = dinv + N_NODES;                       //    65,536 f
    const size_t need = ((size_t)N_NODES * OUT_CH + N_NODES + (size_t)IN_CH * OUT_CH) * 4;
    if (ws_size < need) return;

    float* out = (float*)d_out;

    // 1) transpose W (tiny)
    k_transpose_w<<<(IN_CH * OUT_CH) / 256, 256, 0, stream>>>(W, Wt);

    // 2) GEMM: bufA = x @ W + b   (3125 tiles of 16 rows; 4 waves/block)
    k_gemm_wmma<<<N_NODES / 16, 128, 0, stream>>>(x, Wt, bias, bufA);

    // 3) normalization: deg -> dinv (dinv array doubles as deg accumulator)
    k_deg_init<<<(N_NODES + 255) / 256, 256, 0, stream>>>(dinv);
    k_deg_scatter<<<(N_EDGES + 255) / 256, 256, 0, stream>>>(col, dinv);
    k_dinv<<<(N_NODES + 255) / 256, 256, 0, stream>>>(dinv);

    // 4) three hops, ping-pong bufA <-> d_out, ending in d_out
    const int nElem   = N_NODES * OUT_CH;
    const int gInit   = (nElem + 255) / 256;
    const int gScat   = (N_EDGES + 7) / 8;              // 8 edges (waves) per block

    // hop 1: bufA -> out
    k_hop_init<<<gInit, 256, 0, stream>>>(bufA, dinv, out);
    k_hop_scatter<<<gScat, 256, 0, stream>>>(row, col, dinv, bufA, out);
    // hop 2: out -> bufA
    k_hop_init<<<gInit, 256, 0, stream>>>(out, dinv, bufA);
    k_hop_scatter<<<gScat, 256, 0, stream>>>(row, col, dinv, out, bufA);
    // hop 3: bufA -> out
    k_hop_init<<<gInit, 256, 0, stream>>>(bufA, dinv, out);
    k_hop_scatter<<<gScat, 256, 0, stream>>>(row, col, dinv, bufA, out);
}